// SubtitleAwareLoss_53300544143793
// MI455X (gfx1250) — compile-verified
//
#include <hip/hip_runtime.h>
#include <stdint.h>

typedef __attribute__((ext_vector_type(2))) float v2f;
typedef __attribute__((ext_vector_type(8))) float v8f;

#define NIMG 32
#define HH   640
#define WW   640
#define BOT  448      // bottom_start = int(640*0.7) = 448
#define RH   192      // region height
#define TX   64
#define TY   16
#define GXT  (TX+10)  // 74  gray tile (halo 5)
#define GYT  (TY+10)  // 26
#define MXT  (TX+8)   // 72  bright&contrast (halo 4)
#define MYT  (TY+8)   // 24
#define DXT  (TX+6)   // 70  dilate (halo 3)
#define DYT  (TY+6)   // 22
#define CXT  (TX+4)   // 68  erode -> closing (halo 2)
#define CYT  (TY+4)   // 20
#define EXT  (TX+2)   // 66  erode of closing (halo 1)
#define EYT  (TY+2)   // 18
#define RB   2048     // reduce blocks

__device__ __forceinline__ float max9(const float* b, int i, int st) {
  float v = fmaxf(fmaxf(b[i-st-1], b[i-st]), b[i-st+1]);
  v = fmaxf(v, fmaxf(fmaxf(b[i-1], b[i]), b[i+1]));
  v = fmaxf(v, fmaxf(fmaxf(b[i+st-1], b[i+st]), b[i+st+1]));
  return v;
}
__device__ __forceinline__ float min9(const float* b, int i, int st) {
  float v = fminf(fminf(b[i-st-1], b[i-st]), b[i-st+1]);
  v = fminf(v, fminf(fminf(b[i-1], b[i]), b[i+1]));
  v = fminf(v, fminf(fminf(b[i+st-1], b[i+st]), b[i+st+1]));
  return v;
}

// ---------------------------------------------------------------------------
// Kernel 1: subtitle mask over bottom 192 rows, 16x64 output tile per block.
// Channel halo tiles staged to LDS via CDNA5 async global->LDS copies.
// ---------------------------------------------------------------------------
__global__ __launch_bounds__(256)
void submask_kernel(const float* __restrict__ image, uint8_t* __restrict__ subm) {
  __shared__ float chan[3*GYT*GXT];
  __shared__ float gray[GYT*GXT];
  __shared__ float mbuf[MYT*MXT];
  __shared__ float dbuf[DYT*DXT];
  __shared__ float cbuf[CYT*CXT];
  __shared__ float ebuf[EYT*EXT];

  const int tid = threadIdx.x;
  const int tx0 = blockIdx.x * TX;
  const int ty0 = blockIdx.y * TY;
  const int n   = blockIdx.z;

  // --- Stage 0: async-copy 3 channel halo tiles (reflect-mapped) into LDS ---
  const uint64_t ibase = (uint64_t)(uintptr_t)image;
  for (int idx = tid; idx < 3*GYT*GXT; idx += 256) {
    int ch   = idx / (GYT*GXT);
    int cell = idx - ch*(GYT*GXT);
    int gy = cell / GXT, gx = cell - gy*GXT;
    int ry = ty0 + gy - 5;
    int rx = tx0 + gx - 5;
    // reflect across region boundaries (exact at distance 1, where it matters;
    // farther cells feed positions whose value is overridden below)
    ry = (ry < 0) ? -ry : ((ry >= RH) ? (2*RH - 2 - ry) : ry);
    rx = (rx < 0) ? -rx : ((rx >= WW) ? (2*WW - 2 - rx) : rx);
    int goff = (((n*3 + ch)*HH + (BOT + ry))*WW + rx) * 4;
    uint32_t loff = (uint32_t)(uintptr_t)(&chan[idx]);
    asm volatile("global_load_async_to_lds_b32 %0, %1, %2"
                 :: "v"(loff), "v"(goff), "s"(ibase) : "memory");
  }
  asm volatile("s_wait_asynccnt 0x0" ::: "memory");
  __syncthreads();

  // --- Stage 1: gray = round(0.114 b + 0.587 g + 0.299 r) on clipped bytes ---
  for (int idx = tid; idx < GYT*GXT; idx += 256) {
    float b = fminf(fmaxf(floorf(chan[idx            ]*255.0f), 0.0f), 255.0f);
    float g = fminf(fmaxf(floorf(chan[GYT*GXT   + idx]*255.0f), 0.0f), 255.0f);
    float r = fminf(fmaxf(floorf(chan[2*GYT*GXT + idx]*255.0f), 0.0f), 255.0f);
    gray[idx] = rintf(0.114f*b + 0.587f*g + 0.299f*r);   // RNE, like jnp.round
  }
  __syncthreads();

  // --- Stage 2: m = bright & contrast (halo 4); outside region -> 0 ---
  for (int idx = tid; idx < MYT*MXT; idx += 256) {
    int my = idx / MXT, mx = idx - my*MXT;
    int ry = ty0 + my - 4, rx = tx0 + mx - 4;
    float mval = 0.0f;
    if (ry >= 0 && ry < RH && rx >= 0 && rx < WW) {
      int gi = (my+1)*GXT + (mx+1);
      float gc  = gray[gi];
      float lap = gray[gi-GXT] + gray[gi+GXT] + gray[gi-1] + gray[gi+1] - 4.0f*gc;
      mval = (gc > 204.0f && fabsf(lap) > 76.5f) ? 1.0f : 0.0f;
    }
    mbuf[idx] = mval;
  }
  __syncthreads();

  // --- Stage 3: d = dilate3(m) (halo 3); outside -> 1 (erode pad identity) ---
  for (int idx = tid; idx < DYT*DXT; idx += 256) {
    int dy = idx / DXT, dx = idx - dy*DXT;
    int ry = ty0 + dy - 3, rx = tx0 + dx - 3;
    float v = 1.0f;
    if (ry >= 0 && ry < RH && rx >= 0 && rx < WW)
      v = max9(mbuf, (dy+1)*MXT + (dx+1), MXT);
    dbuf[idx] = v;
  }
  __syncthreads();

  // --- Stage 4: c = erode3(d) = closing (halo 2); outside -> 1 ---
  for (int idx = tid; idx < CYT*CXT; idx += 256) {
    int cy = idx / CXT, cx = idx - cy*CXT;
    int ry = ty0 + cy - 2, rx = tx0 + cx - 2;
    float v = 1.0f;
    if (ry >= 0 && ry < RH && rx >= 0 && rx < WW)
      v = min9(dbuf, (cy+1)*DXT + (cx+1), DXT);
    cbuf[idx] = v;
  }
  __syncthreads();

  // --- Stage 5: e = erode3(c) (halo 1); outside -> 0 (dilate pad identity) ---
  for (int idx = tid; idx < EYT*EXT; idx += 256) {
    int ey = idx / EXT, ex = idx - ey*EXT;
    int ry = ty0 + ey - 1, rx = tx0 + ex - 1;
    float v = 0.0f;
    if (ry >= 0 && ry < RH && rx >= 0 && rx < WW)
      v = min9(cbuf, (ey+1)*CXT + (ex+1), CXT);
    ebuf[idx] = v;
  }
  __syncthreads();

  // --- Stage 6: out = dilate3(e) -> subtitle mask byte ---
  for (int idx = tid; idx < TY*TX; idx += 256) {
    int oy = idx / TX, ox = idx - oy*TX;
    float v = max9(ebuf, (oy+1)*EXT + (ox+1), EXT);
    subm[(size_t)(n*RH + ty0 + oy)*WW + (tx0 + ox)] = (uint8_t)(v > 0.5f);
  }
}

// ---------------------------------------------------------------------------
// Wave-level sum via V_WMMA_F32_16X16X4_F32 with ones-B (layout independent).
// A: lane m holds {acc, 0}; rows of D = acc[m] + acc[m+16] for every column.
// Sum the 8 D registers per lane, then one cross-half add -> 32-lane total.
// ---------------------------------------------------------------------------
__device__ __forceinline__ float wave_sum_wmma(float v) {
  v2f A; A.x = v;    A.y = 0.0f;
  v2f B; B.x = 1.0f; B.y = 1.0f;
  v8f C = {};
  v8f D = __builtin_amdgcn_wmma_f32_16x16x4_f32(false, A, false, B,
                                                (short)0, C, false, false);
  float r = ((D[0]+D[1]) + (D[2]+D[3])) + ((D[4]+D[5]) + (D[6]+D[7]));
  r += __shfl_xor(r, 16, 32);
  return r;
}

// ---------------------------------------------------------------------------
// Kernel 2: streaming BCE + 4 global reductions (float4 per iteration).
// ---------------------------------------------------------------------------
__global__ __launch_bounds__(256)
void reduce_kernel(const float4* __restrict__ pred, const float4* __restrict__ gt,
                   const float4* __restrict__ mask, const uint8_t* __restrict__ subm,
                   float* __restrict__ partial) {
  float a0 = 0.0f, a1 = 0.0f, a2 = 0.0f, a3 = 0.0f;
  const int NQ = NIMG*HH*WW/4;
  for (int q = blockIdx.x*256 + threadIdx.x; q < NQ; q += RB*256) {
    float4 z4 = pred[q], t4 = gt[q], w4 = mask[q];
    int i0  = q*4;
    int n   = i0 / (HH*WW);
    int rem = i0 - n*(HH*WW);
    int y   = rem / WW;
    int x   = rem - y*WW;
    float sb4[4] = {0.f, 0.f, 0.f, 0.f};
    if (y >= BOT) {
      uint32_t sb = *(const uint32_t*)(subm + ((size_t)(n*RH + (y-BOT))*WW + x));
      sb4[0] = (float)( sb        & 0xffu);
      sb4[1] = (float)((sb >>  8) & 0xffu);
      sb4[2] = (float)((sb >> 16) & 0xffu);
      sb4[3] = (float)((sb >> 24) & 0xffu);
    }
    float zb[4] = {z4.x, z4.y, z4.z, z4.w};
    float tb[4] = {t4.x, t4.y, t4.z, t4.w};
    float wb[4] = {w4.x, w4.y, w4.z, w4.w};
#pragma unroll
    for (int k = 0; k < 4; ++k) {
      float z = zb[k], t = tb[k], w = wb[k];
      float bce = fmaxf(z, 0.0f) - z*t + log1pf(__expf(-fabsf(z)));
      float cmb = w * sb4[k];
      a0 += bce * w;
      a1 += w;
      a2 += bce * cmb;
      a3 += cmb;
    }
  }
  // full-EXEC point: wave reductions through the matrix engine
  float r0 = wave_sum_wmma(a0);
  float r1 = wave_sum_wmma(a1);
  float r2 = wave_sum_wmma(a2);
  float r3 = wave_sum_wmma(a3);

  __shared__ float wsum[8*4];
  int wave = threadIdx.x >> 5, lane = threadIdx.x & 31;
  if (lane == 0) {
    wsum[wave*4+0] = r0; wsum[wave*4+1] = r1;
    wsum[wave*4+2] = r2; wsum[wave*4+3] = r3;
  }
  __syncthreads();
  if (threadIdx.x == 0) {
    float s0=0.f, s1=0.f, s2=0.f, s3=0.f;
    for (int w = 0; w < 8; ++w) {
      s0 += wsum[w*4+0]; s1 += wsum[w*4+1];
      s2 += wsum[w*4+2]; s3 += wsum[w*4+3];
    }
    partial[blockIdx.x*4+0] = s0; partial[blockIdx.x*4+1] = s1;
    partial[blockIdx.x*4+2] = s2; partial[blockIdx.x*4+3] = s3;
  }
}

// ---------------------------------------------------------------------------
// Kernel 3: deterministic final reduction + scalar epilogue.
// ---------------------------------------------------------------------------
__global__ __launch_bounds__(256)
void final_kernel(const float* __restrict__ partial, float* __restrict__ out) {
  __shared__ float sm[4][256];
  const int tid = threadIdx.x;
  float s0=0.f, s1=0.f, s2=0.f, s3=0.f;
  for (int i = tid; i < RB; i += 256) {
    s0 += partial[4*i+0]; s1 += partial[4*i+1];
    s2 += partial[4*i+2]; s3 += partial[4*i+3];
  }
  sm[0][tid]=s0; sm[1][tid]=s1; sm[2][tid]=s2; sm[3][tid]=s3;
  __syncthreads();
  for (int st = 128; st > 0; st >>= 1) {
    if (tid < st) {
      sm[0][tid] += sm[0][tid+st]; sm[1][tid] += sm[1][tid+st];
      sm[2][tid] += sm[2][tid+st]; sm[3][tid] += sm[3][tid+st];
    }
    __syncthreads();
  }
  if (tid == 0) {
    float Sbm = sm[0][0], Sm = sm[1][0], Sbc = sm[2][0], Sc = sm[3][0];
    float base_loss = Sbm / fmaxf(Sm, 1e-6f);
    float sub_loss  = Sbc / fmaxf(Sc, 1e-6f);
    out[0] = (Sc > 0.0f) ? sub_loss : base_loss;
  }
}

// ---------------------------------------------------------------------------
extern "C" void kernel_launch(void* const* d_in, const int* in_sizes, int n_in,
                              void* d_out, int out_size, void* d_ws, size_t ws_size,
                              hipStream_t stream) {
  const float* pred  = (const float*)d_in[0];
  const float* gt    = (const float*)d_in[1];
  const float* mask  = (const float*)d_in[2];
  const float* image = (const float*)d_in[3];

  uint8_t* subm   = (uint8_t*)d_ws;                              // 32*192*640 bytes
  float*   partial = (float*)((char*)d_ws + (size_t)NIMG*RH*WW); // RB*4 floats

  dim3 g1(WW/TX, RH/TY, NIMG);   // 10 x 12 x 32 tiles
  submask_kernel<<<g1, 256, 0, stream>>>(image, subm);
  reduce_kernel<<<RB, 256, 0, stream>>>((const float4*)pred, (const float4*)gt,
                                        (const float4*)mask, subm, partial);
  final_kernel<<<1, 256, 0, stream>>>(partial, (float*)d_out);
}